// LocalSparseAttention_90640989815087
// MI455X (gfx1250) — compile-verified
//
#include <hip/hip_runtime.h>

typedef __attribute__((ext_vector_type(2))) float v2f;
typedef __attribute__((ext_vector_type(4))) float v4f;
typedef __attribute__((ext_vector_type(8))) float v8f;

#define BB 2
#define HH 16
#define TT 2048
#define DD 64
#define WW 128
#define NKT 17            // key tiles per query tile: window 257 -> 17*16 = 272 cols
#define NCOL (NKT * 16)   // 272
#define LDS_STRIDE 274    // even -> 8B-aligned float2 rows
#define WPB 2             // waves (query tiles) per block

// ---------------------------------------------------------------------------
// Complementary zero-fill: for each 16-row query tile the attention kernel
// writes attn cols [max(0,q0-128), min(T,q0+144)) (tile-aligned). Zero exactly
// the rest with non-temporal 128-bit stores — no byte written twice, and the
// ~466 MB of zeros bypass L2 (which we want to keep for K/V).
// ---------------------------------------------------------------------------
__global__ __launch_bounds__(256)
void lsa_zero_kernel(float* __restrict__ attn) {
    const int gw = blockIdx.x;              // tile id: bh*128 + qt
    const int qt = gw & (TT / 16 - 1);
    const int bh = gw >> 7;
    const int q0 = qt * 16;

    int L = q0 - WW;       if (L < 0)  L = 0;        // window start (mult of 16)
    int R = q0 + WW + 16;  if (R > TT) R = TT;       // window end   (mult of 16)
    const int l4 = L >> 2;                           // float4 count, left span
    const int r4 = (TT - R) >> 2;                    // float4 count, right span

    const v4f z = {0.f, 0.f, 0.f, 0.f};
    const size_t rowbase = ((size_t)bh * TT + q0) * TT;

    for (int m = 0; m < 16; ++m) {
        v4f* rowL = (v4f*)(attn + rowbase + (size_t)m * TT);
        for (int c = threadIdx.x; c < l4; c += 256)
            __builtin_nontemporal_store(z, rowL + c);
        v4f* rowR = (v4f*)(attn + rowbase + (size_t)m * TT + R);
        for (int c = threadIdx.x; c < r4; c += 256)
            __builtin_nontemporal_store(z, rowR + c);
    }
}

// ---------------------------------------------------------------------------
// Local sparse attention: one wave per 16-query tile
// ---------------------------------------------------------------------------
__global__ __launch_bounds__(WPB * 32)
void lsa_attn_kernel(const float* __restrict__ Q,
                     const float* __restrict__ K,
                     const float* __restrict__ V,
                     float* __restrict__ Out,
                     float* __restrict__ Attn) {
    __shared__ float s_p[WPB][16][LDS_STRIDE];
    __shared__ float s_inv[WPB][16];

    const int lane = threadIdx.x & 31;
    const int wave = threadIdx.x >> 5;
    const int gw   = blockIdx.x * WPB + wave;       // global wave id
    const int qt   = gw & (TT / 16 - 1);            // 128 query tiles per (b,h)
    const int bh   = gw >> 7;
    const int q0   = qt * 16;
    const int half = lane >> 4;                     // 0 or 1
    const int l15  = lane & 15;

    const size_t base = (size_t)bh * TT * DD;

    // ---- Q tile in A-fragment layout: lane holds row M=l15, K = 4c + 2*half + {0,1}
    const float* qrow = Q + base + (size_t)(q0 + l15) * DD + 2 * half;
    v2f a[16];
#pragma unroll
    for (int c = 0; c < 16; ++c)
        a[c] = *(const v2f*)(qrow + 4 * c);

    const v2f vzero = {0.f, 0.f};

    // ---- S = (Q K^T) * scale, masked, staged in LDS -------------------------
    for (int t = 0; t < NKT; ++t) {
        const int  kt     = q0 - WW + t * 16;
        const int  krow   = kt + l15;                 // key row this lane supplies (B layout)
        const bool kvalid = (krow >= 0) && (krow < TT);
        const float* kptr = K + base + (size_t)(kvalid ? krow : 0) * DD + 2 * half;

        v8f acc = {0.f, 0.f, 0.f, 0.f, 0.f, 0.f, 0.f, 0.f};
#pragma unroll
        for (int c = 0; c < 16; ++c) {
            v2f b = kvalid ? *(const v2f*)(kptr + 4 * c) : vzero;
            acc = __builtin_amdgcn_wmma_f32_16x16x4_f32(
                false, a[c], false, b, (short)0, acc, false, false);
        }
        // C/D layout: VGPR r -> row M = r + 8*half, col N = l15
#pragma unroll
        for (int r = 0; r < 8; ++r) {
            const int m  = r + 8 * half;
            const int qg = q0 + m;
            const int kg = kt + l15;
            float s = acc[r] * 0.125f;                 // 1/sqrt(64)
            int dist = qg - kg; dist = dist < 0 ? -dist : dist;
            if (kg < 0 || kg >= TT || dist > WW) s = -1e9f;
            s_p[wave][m][t * 16 + l15] = s;
        }
    }
    __syncthreads();

    // ---- row softmax: 2 lanes per row, each scans 136 columns ---------------
    {
        const int c0 = half * (NCOL / 2);
        float mx = -3.4e38f;
        for (int c = 0; c < NCOL / 2; ++c)
            mx = fmaxf(mx, s_p[wave][l15][c0 + c]);
        mx = fmaxf(mx, __shfl_xor(mx, 16, 32));
        float sum = 0.f;
        for (int c = 0; c < NCOL / 2; ++c) {
            float e = __expf(s_p[wave][l15][c0 + c] - mx);
            s_p[wave][l15][c0 + c] = e;
            sum += e;
        }
        sum += __shfl_xor(sum, 16, 32);
        if (half == 0) s_inv[wave][l15] = 1.0f / sum;
    }
    __syncthreads();

    // ---- write normalized attn: coalesced 64B row segments, non-temporal ----
#pragma unroll 1
    for (int t = 0; t < NKT; ++t) {
        const int kt = q0 - WW + t * 16;
        const int kg = kt + l15;
        if (kg >= 0 && kg < TT) {
#pragma unroll
            for (int r = 0; r < 8; ++r) {
                const int m = r + 8 * half;
                const float pnorm = s_p[wave][m][t * 16 + l15] * s_inv[wave][m];
                __builtin_nontemporal_store(
                    pnorm, &Attn[((size_t)bh * TT + (q0 + m)) * TT + kg]);
            }
        }
    }

    // ---- O = P * V ----------------------------------------------------------
    const float inv_m = s_inv[wave][l15];             // A rows: M = l15 for all lanes
    const float* vbase = V + base;
#pragma unroll 1
    for (int nc = 0; nc < 4; ++nc) {
        v8f acc = {0.f, 0.f, 0.f, 0.f, 0.f, 0.f, 0.f, 0.f};
        const int ncol = nc * 16 + l15;
        for (int kc = 0; kc < NCOL / 4; ++kc) {       // 68 WMMAs over K = 272
            const int c0 = kc * 4 + 2 * half;         // A: P[m][c0..c0+1]
            v2f pa = *(const v2f*)(&s_p[wave][l15][c0]);
            pa = pa * inv_m;                          // fold softmax normalization into A
            const int kr = q0 - WW + c0;              // B: V rows kr, kr+1, col = ncol
            float b0 = (kr     >= 0 && kr     < TT) ? vbase[(size_t)kr       * DD + ncol] : 0.f;
            float b1 = (kr + 1 >= 0 && kr + 1 < TT) ? vbase[(size_t)(kr + 1) * DD + ncol] : 0.f;
            v2f pb = {b0, b1};
            acc = __builtin_amdgcn_wmma_f32_16x16x4_f32(
                false, pa, false, pb, (short)0, acc, false, false);
        }
#pragma unroll
        for (int r = 0; r < 8; ++r) {
            const int m = r + 8 * half;
            Out[((size_t)bh * TT + (q0 + m)) * DD + nc * 16 + l15] = acc[r];
        }
    }
}

// ---------------------------------------------------------------------------
extern "C" void kernel_launch(void* const* d_in, const int* in_sizes, int n_in,
                              void* d_out, int out_size, void* d_ws, size_t ws_size,
                              hipStream_t stream) {
    const float* q = (const float*)d_in[0];
    const float* k = (const float*)d_in[1];
    const float* v = (const float*)d_in[2];

    float* out  = (float*)d_out;
    const size_t out_elems = (size_t)BB * HH * TT * DD;   // 4,194,304
    float* attn = out + out_elems;

    const int ntiles = BB * HH * (TT / 16);               // 4096

    // 1) zero exactly the out-of-window attn region (non-temporal)
    lsa_zero_kernel<<<ntiles, 256, 0, stream>>>(attn);

    // 2) attention: 4096 waves, WPB waves per block
    lsa_attn_kernel<<<ntiles / WPB, WPB * 32, 0, stream>>>(q, k, v, out, attn);
}